// HeteroModel_9895604650660
// MI455X (gfx1250) — compile-verified
//
#include <hip/hip_runtime.h>
#include <hip/hip_bf16.h>

typedef __attribute__((ext_vector_type(2))) float v2f;
typedef __attribute__((ext_vector_type(8))) float v8f;

// Problem constants (match reference)
constexpr int    kN   = 100000;
constexpr int    kE   = 400000;
constexpr int    kR   = 3;
constexpr int    kH   = 128;      // IN = HID = OUT = 128
constexpr int    kKC  = 512;      // concat K = 128 * (1 + R)

// Workspace layout (in floats)
constexpr size_t AGG_OFF   = 0;                                  // [R,N,128]
constexpr size_t AGG_CNT   = (size_t)kR * kN * kH;               // 38,400,000
constexpr size_t CNT_OFF   = AGG_OFF + AGG_CNT;                  // [R,N]
constexpr size_t CNT_CNT   = (size_t)kR * kN;
constexpr size_t H1_OFF    = CNT_OFF + CNT_CNT;                  // [N,128]
constexpr size_t H2_OFF    = H1_OFF + (size_t)kN * kH;           // [N,128]
constexpr size_t WCAT1_OFF = H2_OFF + (size_t)kN * kH;           // [512,128]
constexpr size_t BS1_OFF   = WCAT1_OFF + (size_t)kKC * kH;       // [128]
constexpr size_t WCAT2_OFF = BS1_OFF + kH;                       // [512,128]
constexpr size_t BS2_OFF   = WCAT2_OFF + (size_t)kKC * kH;       // [128]

// ---------------------------------------------------------------- zero
__global__ __launch_bounds__(256)
void zero_kernel(float* __restrict__ p, size_t n) {
    size_t gid = (size_t)blockIdx.x * 256 + threadIdx.x;
    if (gid < n) p[gid] = 0.0f;
}

// ------------------------------------------------ build W_cat & bias sum
// Wcat[512][128]: rows 0..127 = sum_r Wself[r], rows 128..511 = Wneigh (R,128,128)
__global__ __launch_bounds__(256)
void weights_kernel(const float* __restrict__ Wself, const float* __restrict__ Wneigh,
                    const float* __restrict__ b,
                    float* __restrict__ Wcat, float* __restrict__ bsum) {
    int gid = blockIdx.x * 256 + threadIdx.x;         // 0 .. 512*128-1
    if (gid < kKC * kH) {
        int k = gid >> 7;
        int n = gid & 127;
        float v;
        if (k < kH) {
            v = Wself[k * kH + n] + Wself[kH * kH + k * kH + n] + Wself[2 * kH * kH + k * kH + n];
        } else {
            v = Wneigh[(size_t)(k - kH) * kH + n];    // contiguous (R,128,128)
        }
        Wcat[gid] = v;
    }
    if (gid < kH) bsum[gid] = b[gid] + b[kH + gid] + b[2 * kH + gid];
}

// ---------------------------------------------------------- aggregation
// one 128-float row per edge; thread t of edge handles feature t
__global__ __launch_bounds__(256)
void aggregate_kernel(const float* __restrict__ h, const int* __restrict__ src,
                      const int* __restrict__ dst, float* __restrict__ agg,
                      float* __restrict__ cnt, int do_cnt) {
    size_t gid  = (size_t)blockIdx.x * 256 + threadIdx.x;
    size_t edge = gid >> 7;                            // r*E + e
    int    t    = (int)(gid & 127);
    if (edge >= (size_t)kR * kE) return;
    int s = src[edge];
    int d = dst[edge];
    size_t r = edge / kE;
    atomicAdd(&agg[((size_t)r * kN + (size_t)d) * kH + t], h[(size_t)s * kH + t]);
    if (do_cnt && t == 0) atomicAdd(&cnt[(size_t)r * kN + d], 1.0f);
}

// ------------------------------------------------------------- sum->mean
__global__ __launch_bounds__(256)
void mean_kernel(float* __restrict__ agg, const float* __restrict__ cnt) {
    size_t gid = (size_t)blockIdx.x * 256 + threadIdx.x;
    if (gid < AGG_CNT) {
        float c = cnt[gid >> 7];
        agg[gid] = agg[gid] / fmaxf(c, 1.0f);
    }
}

// ------------------------------------------------------- WMMA fp32 GEMM
// out[N,128] = act( [hin | mean0 | mean1 | mean2] @ Wcat + bsum )
// block = 256 threads (8 waves); block computes 32 rows x 128 cols.
// Each wave owns a 16-col strip and two stacked 16x16 accumulators so that
// every B fragment load feeds TWO v_wmma_f32_16x16x4_f32 ops.
constexpr int ROWSTRIDE = 516;   // pad 512 -> 516 floats: conflict-free ds_load_b64

__global__ __launch_bounds__(256)
void gemm_kernel(const float* __restrict__ hin, const float* __restrict__ mean,
                 const float* __restrict__ Wcat, const float* __restrict__ bsum,
                 float* __restrict__ out, int relu) {
    __shared__ float Atile[32 * ROWSTRIDE];            // 66,048 B
    const int tid = threadIdx.x;
    const int r0  = blockIdx.x * 32;                   // N = 3125*32 exactly

    // Stage A tile: 32 rows x 512 cols from 4 sources (16384 elems, 64/thread)
    #pragma unroll 4
    for (int i = 0; i < 64; ++i) {
        int f   = tid + 256 * i;                       // 0..16383
        int row = f >> 9;
        int col = f & 511;
        int seg = col >> 7;
        int lc  = col & 127;
        float v;
        if (seg == 0)
            v = hin[(size_t)(r0 + row) * kH + lc];
        else
            v = mean[((size_t)(seg - 1) * kN + (r0 + row)) * kH + lc];
        Atile[row * ROWSTRIDE + col] = v;
    }
    __syncthreads();

    const int lane  = tid & 31;
    const int wave  = tid >> 5;
    const int ml    = lane & 15;                       // A row / B col within tile
    const int khalf = lane >> 4;                       // 0: K{0,1}, 1: K{2,3}
    const int nc    = wave * 16 + ml;                  // global output column

    float bias = bsum[nc];
    v8f acc0, acc1;
    #pragma unroll
    for (int i = 0; i < 8; ++i) { acc0[i] = bias; acc1[i] = bias; }

    for (int k0 = 0; k0 < kKC; k0 += 4) {
        const int kb = k0 + 2 * khalf;
        v2f a0, a1;
        a0.x = Atile[ml * ROWSTRIDE + kb];
        a0.y = Atile[ml * ROWSTRIDE + kb + 1];
        a1.x = Atile[(ml + 16) * ROWSTRIDE + kb];
        a1.y = Atile[(ml + 16) * ROWSTRIDE + kb + 1];
        v2f b;
        b.x = Wcat[(size_t)kb * kH + nc];
        b.y = Wcat[(size_t)(kb + 1) * kH + nc];
        acc0 = __builtin_amdgcn_wmma_f32_16x16x4_f32(false, a0, false, b,
                                                     (short)0, acc0, false, false);
        acc1 = __builtin_amdgcn_wmma_f32_16x16x4_f32(false, a1, false, b,
                                                     (short)0, acc1, false, false);
    }

    #pragma unroll
    for (int i = 0; i < 8; ++i) {
        const int rowL = i + 8 * khalf;                // C layout: lanes>=16 hold M=i+8
        float v0 = acc0[i];
        float v1 = acc1[i];
        if (relu) {
            v0 = v0 > 0.0f ? v0 : 0.0f;
            v1 = v1 > 0.0f ? v1 : 0.0f;
        }
        out[(size_t)(r0 + rowL) * kH + nc]      = v0;
        out[(size_t)(r0 + 16 + rowL) * kH + nc] = v1;
    }
}

// ----------------------------------------------------------- edge scores
// one wave per edge; 32 lanes x float4 = 128 features; wave32 xor-reduction
__global__ __launch_bounds__(256)
void score_kernel(const float* __restrict__ h,
                  const int* __restrict__ pos_src, const int* __restrict__ pos_dst,
                  const int* __restrict__ neg_src, const int* __restrict__ neg_dst,
                  const int* __restrict__ relation, float* __restrict__ out) {
    const int lane = threadIdx.x & 31;
    const int wave = threadIdx.x >> 5;
    const size_t ew = (size_t)blockIdx.x * 8 + wave;
    if (ew >= 2 * (size_t)kE) return;
    const int  rel = relation[0];
    const bool neg = ew >= (size_t)kE;
    const size_t e = neg ? ew - kE : ew;
    const int* sp = neg ? neg_src : pos_src;
    const int* dp = neg ? neg_dst : pos_dst;
    const int s = sp[(size_t)rel * kE + e];
    const int d = dp[(size_t)rel * kE + e];
    const float4 a = reinterpret_cast<const float4*>(h + (size_t)s * kH)[lane];
    const float4 b = reinterpret_cast<const float4*>(h + (size_t)d * kH)[lane];
    float sum = a.x * b.x + a.y * b.y + a.z * b.z + a.w * b.w;
    #pragma unroll
    for (int off = 16; off > 0; off >>= 1) sum += __shfl_xor(sum, off, 32);
    if (lane == 0) out[ew] = sum;
}

// ---------------------------------------------------------------- driver
extern "C" void kernel_launch(void* const* d_in, const int* in_sizes, int n_in,
                              void* d_out, int out_size, void* d_ws, size_t ws_size,
                              hipStream_t stream) {
    (void)in_sizes; (void)n_in; (void)out_size; (void)ws_size;

    const float* x        = (const float*)d_in[0];
    const int*   pos_src  = (const int*)d_in[1];
    const int*   pos_dst  = (const int*)d_in[2];
    const int*   neg_src  = (const int*)d_in[3];
    const int*   neg_dst  = (const int*)d_in[4];
    const float* W1_self  = (const float*)d_in[5];
    const float* W1_neigh = (const float*)d_in[6];
    const float* b1       = (const float*)d_in[7];
    const float* W2_self  = (const float*)d_in[8];
    const float* W2_neigh = (const float*)d_in[9];
    const float* b2       = (const float*)d_in[10];
    const int*   relation = (const int*)d_in[11];

    float* ws    = (float*)d_ws;
    float* agg   = ws + AGG_OFF;
    float* cnt   = ws + CNT_OFF;
    float* H1    = ws + H1_OFF;
    float* H2    = ws + H2_OFF;
    float* Wcat1 = ws + WCAT1_OFF;
    float* bs1   = ws + BS1_OFF;
    float* Wcat2 = ws + WCAT2_OFF;
    float* bs2   = ws + BS2_OFF;
    float* scores = (float*)d_out;

    auto blocks = [](size_t n) { return (unsigned)((n + 255) / 256); };

    // weights (both layers)
    weights_kernel<<<blocks((size_t)kKC * kH), 256, 0, stream>>>(W1_self, W1_neigh, b1, Wcat1, bs1);
    weights_kernel<<<blocks((size_t)kKC * kH), 256, 0, stream>>>(W2_self, W2_neigh, b2, Wcat2, bs2);

    // ---- layer 1
    zero_kernel<<<blocks(AGG_CNT + CNT_CNT), 256, 0, stream>>>(agg, AGG_CNT + CNT_CNT); // agg + cnt
    aggregate_kernel<<<blocks((size_t)kR * kE * kH), 256, 0, stream>>>(x, pos_src, pos_dst, agg, cnt, 1);
    mean_kernel<<<blocks(AGG_CNT), 256, 0, stream>>>(agg, cnt);
    gemm_kernel<<<kN / 32, 256, 0, stream>>>(x, agg, Wcat1, bs1, H1, 1);

    // ---- layer 2 (counts unchanged; re-zero sums only)
    zero_kernel<<<blocks(AGG_CNT), 256, 0, stream>>>(agg, AGG_CNT);
    aggregate_kernel<<<blocks((size_t)kR * kE * kH), 256, 0, stream>>>(H1, pos_src, pos_dst, agg, cnt, 0);
    mean_kernel<<<blocks(AGG_CNT), 256, 0, stream>>>(agg, cnt);
    gemm_kernel<<<kN / 32, 256, 0, stream>>>(H1, agg, Wcat2, bs2, H2, 0);

    // ---- scores (pos then neg, concatenated)
    score_kernel<<<blocks(2 * (size_t)kE * 32), 256, 0, stream>>>(
        H2, pos_src, pos_dst, neg_src, neg_dst, relation, scores);
}